// VectorQuantizer_60438779789623
// MI455X (gfx1250) — compile-verified
//
#include <hip/hip_runtime.h>

typedef float v2f __attribute__((ext_vector_type(2)));
typedef float v8f __attribute__((ext_vector_type(8)));

#define CB_K   512   // codebook entries
#define EMB_E  64    // embedding dim
#define HW     1024  // 32*32 spatial positions per image

// One wave32 handles 32 query vectors (two 16-row WMMA tiles sharing every
// B fragment) against all 512 codes with V_WMMA_F32_16X16X4_F32.
// argmin(||x||^2 - 2 x.c + ||c||^2)  ==  argmax(x.c - ||c||^2/2); the
// -||c||^2/2 term is pre-negated/halved in LDS and folded into the WMMA
// C-matrix initializer, and its LDS load is software-pipelined one k-tile
// ahead so the hot loop has no ds-wait bubble.
__global__ __launch_bounds__(256) void vq_wmma_f32_kernel(
    const float* __restrict__ emb,   // [B, 64, 32, 32]
    const float* __restrict__ cb,    // [512, 64]
    float* __restrict__ out)         // [B, 64, 32, 32]
{
  __shared__ float s_ci[CB_K + 16]; // -||c||^2/2 per code (+pad for prefetch)
  __shared__ int   s_idx[8 * 32];   // winning code per query row

  const int lane = threadIdx.x & 31;
  const int wv   = threadIdx.x >> 5;       // wave in block (0..7)
  const int w    = blockIdx.x * 8 + wv;    // global 32-row tile id
  const int lm   = lane & 15;
  const int half = lane >> 4;

  const int b       = w >> 5;              // 1024/32 = 32 tiles per image
  const int hw_base = (w & 31) << 5;

  // ---- issue A loads first: 32 rows (n) x 64 (e), WMMA 16x4 f32 A layout ----
  const float* embp = emb + (size_t)b * (EMB_E * HW) + hw_base + lm;
  v2f a0[16], a1[16];
#pragma unroll
  for (int t = 0; t < 16; ++t) {
    const int e0 = 4 * t + 2 * half;
    a0[t].x = embp[(size_t)e0 * HW];
    a0[t].y = embp[(size_t)(e0 + 1) * HW];
    a1[t].x = embp[(size_t)e0 * HW + 16];
    a1[t].y = embp[(size_t)(e0 + 1) * HW + 16];
  }

  // ---- precompute -||c||^2/2 for all 512 codes, once per block ----
  for (int r = threadIdx.x; r < CB_K; r += 256) {
    const float4* row = (const float4*)(cb + (size_t)r * EMB_E);
    float s = 0.f;
#pragma unroll
    for (int q = 0; q < 16; ++q) {
      const float4 vv = row[q];
      s += vv.x * vv.x + vv.y * vv.y + vv.z * vv.z + vv.w * vv.w;
    }
    s_ci[r] = -0.5f * s;
  }
  if (threadIdx.x < 16) s_ci[CB_K + threadIdx.x] = 0.f;  // prefetch pad

  __syncthreads();                         // s_ci ready

  float mv0[8], mv1[8];
  int   mi0[8], mi1[8];
#pragma unroll
  for (int v = 0; v < 8; ++v) {
    mv0[v] = mv1[v] = -3.4e38f;
    mi0[v] = mi1[v] = 0;
  }

  // B layout: lane holds N = lm (code); VGPR v + 2*half selects K (e).
  const float2* cbp = (const float2*)(cb + (size_t)lm * EMB_E + 2 * half);

  float ci = s_ci[lm];                     // C-init for k-tile 0
  for (int kt = 0; kt < CB_K / 16; ++kt) {
    const int   code    = kt * 16 + lm;
    const float ci_next = s_ci[code + 16]; // prefetch next tile's C-init
    v8f acc0 = {ci, ci, ci, ci, ci, ci, ci, ci};
    v8f acc1 = acc0;
#pragma unroll
    for (int t = 0; t < 16; ++t) {
      const float2 bl = cbp[2 * t];
      v2f bv; bv.x = bl.x; bv.y = bl.y;
      acc0 = __builtin_amdgcn_wmma_f32_16x16x4_f32(
          false, a0[t], false, bv, (short)0, acc0, false, false);
      acc1 = __builtin_amdgcn_wmma_f32_16x16x4_f32(
          false, a1[t], false, bv, (short)0, acc1, false, false);
    }
    // D layout: VGPR v, lane -> row n = v + 8*half (+16 for acc1), col = lm
#pragma unroll
    for (int v = 0; v < 8; ++v) {
      if (acc0[v] > mv0[v]) { mv0[v] = acc0[v]; mi0[v] = code; }
      if (acc1[v] > mv1[v]) { mv1[v] = acc1[v]; mi1[v] = code; }
    }
    cbp += 16 * EMB_E / 2;                 // next 16 codebook rows
    ci = ci_next;
  }

  // ---- cross-lane argmax within each 16-lane half (butterfly) ----
#pragma unroll
  for (int v = 0; v < 8; ++v) {
    float m0 = mv0[v], m1 = mv1[v];
    int   i0 = mi0[v], i1 = mi1[v];
#pragma unroll
    for (int mask = 1; mask <= 8; mask <<= 1) {
      const float o0 = __shfl_xor(m0, mask, 32);
      const int   q0 = __shfl_xor(i0, mask, 32);
      const float o1 = __shfl_xor(m1, mask, 32);
      const int   q1 = __shfl_xor(i1, mask, 32);
      if (o0 > m0 || (o0 == m0 && q0 < i0)) { m0 = o0; i0 = q0; }
      if (o1 > m1 || (o1 == m1 && q1 < i1)) { m1 = o1; i1 = q1; }
    }
    if (lm == 0) {
      s_idx[wv * 32 + half * 8 + v]      = i0;   // rows 0..15
      s_idx[wv * 32 + 16 + half * 8 + v] = i1;   // rows 16..31
    }
  }
  __syncthreads();

  // ---- decode: out[b, e, hw_base + lane] = codebook[idx[n], e] ----
  const int     nidx  = s_idx[wv * 32 + lane];
  const float4* cbrow = (const float4*)(cb + (size_t)nidx * EMB_E);
  float*        outp  = out + (size_t)b * (EMB_E * HW) + hw_base + lane;
#pragma unroll 4
  for (int eq = 0; eq < EMB_E / 4; ++eq) {
    const float4 vv = cbrow[eq];           // b128 gather of this row chunk
    outp[(size_t)(4 * eq + 0) * HW] = vv.x;
    outp[(size_t)(4 * eq + 1) * HW] = vv.y;
    outp[(size_t)(4 * eq + 2) * HW] = vv.z;
    outp[(size_t)(4 * eq + 3) * HW] = vv.w;
  }
}

extern "C" void kernel_launch(void* const* d_in, const int* in_sizes, int n_in,
                              void* d_out, int out_size, void* d_ws, size_t ws_size,
                              hipStream_t stream) {
  const float* emb = (const float*)d_in[0];   // [B, 64, 32, 32] f32
  const float* cb  = (const float*)d_in[1];   // [512, 64] f32
  float*       out = (float*)d_out;           // [B, 64, 32, 32] f32
  (void)d_ws; (void)ws_size; (void)n_in; (void)out_size;

  const int B = in_sizes[0] / (EMB_E * HW);   // 64 for the reference shapes
  // waves total = B * (HW/32) = B*32; 8 waves (256 threads) per block
  const int blocks = B * 4;
  vq_wmma_f32_kernel<<<blocks, 256, 0, stream>>>(emb, cb, out);
}